// ProposalFilter_63264868270541
// MI455X (gfx1250) — compile-verified
//
#include <hip/hip_runtime.h>
#include <stdint.h>

// ---------------- problem constants (match reference) ----------------
#define NMS_B        32
#define NMS_N        100000
#define NMS_K        200
#define NMS_THRESH   0.5f
#define NMS_MINAREA  4.0f

// candidate budget per image: exact radix-select of top TARGETC keys,
// accept-all >= threshold gives <= SORTN with overwhelming probability.
#define SORTN        8192
#define TARGETC      (SORTN - 8)
#define NTHREADS     1024
#define NWAVES       (NTHREADS / 32)     // wave32 on gfx1250
#define HPAD         257                 // 257-entry stride decorrelates LDS banks

#define USE_ASYNC_GATHER 1

// ---------------- dynamic LDS layout for the NMS kernel (bytes) ------
#define SM_KEY   0                          // uint64 [SORTN]      64 KB
#define SM_IDX   (SORTN * 8)                // uint32 [SORTN]      32 KB
#define SM_BOX   (SM_IDX + SORTN * 4)       // float  [SORTN*4]   128 KB (AoS y1,x1,y2,x2)
#define SM_SUP   (SM_BOX + SORTN * 16)      // uint32 [SORTN/32]    1 KB
#define SM_KEEP  (SM_SUP + SORTN / 8)       // int    [NMS_K]
#define SM_CTL   (SM_KEEP + NMS_K * 4)      // int    [4] : 0=cur 1=ptr 2=kept
#define SM_TOTAL (SM_CTL + 16)

// =====================================================================
// Kernel 1: sortable key per box.
// softmax(s)[0] = sigmoid(s0 - s1) is monotone in d = s0 - s1, so sorting
// by the monotone-uint encoding of d reproduces argsort(-probs).
// =====================================================================
__global__ void score_key_kernel(const float2* __restrict__ scores,
                                 uint32_t* __restrict__ ukeys, int total) {
    int i = blockIdx.x * blockDim.x + threadIdx.x;
    if (i >= total) return;
    float2 s = scores[i];
    float d = s.x - s.y;
    uint32_t b = __float_as_uint(d);
    // monotone map: larger float -> larger uint
    ukeys[i] = (b & 0x80000000u) ? ~b : (b | 0x80000000u);
}

// =====================================================================
// Kernel 2: per-image 4-pass radix-select (MSB->LSB bytes) of the exact
// value of the TARGETC-th largest key. One block per image.
// Per-wave privatized histograms: exponent clustering would otherwise
// serialize ~100K same-address LDS RMWs per pass.
// =====================================================================
__global__ __launch_bounds__(NTHREADS)
void radix_select_kernel(const uint32_t* __restrict__ ukeys,
                         uint32_t* __restrict__ thr,
                         uint32_t* __restrict__ selCount) {
    __shared__ uint32_t whist[NWAVES * HPAD];   // per-wave 256-bin copies (+pad)
    __shared__ uint32_t hsum[256];
    __shared__ uint32_t sh_prefix;
    __shared__ int      sh_need;

    const int b   = blockIdx.x;
    const int tid = threadIdx.x;
    const int wid = tid >> 5;                   // wave32
    const uint32_t* ub = ukeys + (size_t)b * NMS_N;
    if (tid == 0) { sh_prefix = 0u; sh_need = TARGETC; }

    for (int pass = 0; pass < 4; ++pass) {
        for (int i = tid; i < NWAVES * HPAD; i += NTHREADS) whist[i] = 0u;
        __syncthreads();
        const uint32_t prefix = sh_prefix;
        const int shift = 24 - 8 * pass;
        uint32_t* myh = &whist[wid * HPAD];
        for (int j = tid; j < NMS_N; j += NTHREADS) {
            uint32_t uu = ub[j];
            bool match = (pass == 0) ||
                         ((uu >> (shift + 8)) == (prefix >> (shift + 8)));
            if (match) atomicAdd(&myh[(uu >> shift) & 255u], 1u);
        }
        __syncthreads();
        if (tid < 256) {
            uint32_t s = 0u;
            for (int w = 0; w < NWAVES; ++w) s += whist[w * HPAD + tid];
            hsum[tid] = s;
        }
        __syncthreads();
        if (tid == 0) {
            int need = sh_need;
            uint32_t cum = 0u;
            int sel = 0;
            for (int bin = 255; bin >= 0; --bin) {
                uint32_t c = hsum[bin];
                if (cum + c >= (uint32_t)need) { sel = bin; sh_need = need - (int)cum; break; }
                cum += c;
            }
            sh_prefix = prefix | ((uint32_t)sel << shift);
        }
        __syncthreads();
    }
    if (tid == 0) {
        thr[b] = sh_prefix;      // exact TARGETC-th largest key
        selCount[b] = 0u;        // reset for compaction (fresh every call)
    }
}

// =====================================================================
// Kernel 3: compact candidates with key >= threshold (order fixed later
// by the deterministic 64-bit sort key, so atomic order doesn't matter).
// =====================================================================
__global__ void compact_kernel(const uint32_t* __restrict__ ukeys,
                               const uint32_t* __restrict__ thr,
                               uint32_t* __restrict__ selCount,
                               uint32_t* __restrict__ candU,
                               uint32_t* __restrict__ candIdx) {
    int i = blockIdx.x * blockDim.x + threadIdx.x;
    if (i >= NMS_B * NMS_N) return;
    int b = i / NMS_N;
    int j = i - b * NMS_N;
    uint32_t uu = ukeys[i];
    if (uu >= thr[b]) {
        uint32_t pos = atomicAdd(&selCount[b], 1u);
        if (pos < SORTN) {
            candU[(size_t)b * SORTN + pos]   = uu;
            candIdx[(size_t)b * SORTN + pos] = (uint32_t)j;
        }
    }
}

// =====================================================================
// Kernel 4: per-image bitonic sort (64-bit key = score | ~idx, stable
// like jnp.argsort), async LDS gather of candidate boxes, greedy NMS,
// output write. One 1024-thread block per image, ~226 KB dynamic LDS.
// =====================================================================
__global__ __launch_bounds__(NTHREADS)
void nms_kernel(const float* __restrict__ bbox,
                const uint32_t* __restrict__ candU,
                const uint32_t* __restrict__ candIdx,
                const uint32_t* __restrict__ selCount,
                float* __restrict__ out) {
    extern __shared__ unsigned char smem[];
    uint64_t* skey = (uint64_t*)(smem + SM_KEY);
    uint32_t* sidx = (uint32_t*)(smem + SM_IDX);
    float*    sbox = (float*)   (smem + SM_BOX);
    uint32_t* ssup = (uint32_t*)(smem + SM_SUP);
    int*      skeep = (int*)    (smem + SM_KEEP);
    int*      sctl  = (int*)    (smem + SM_CTL);

    const int b   = blockIdx.x;
    const int tid = threadIdx.x;
    uint32_t scount = selCount[b];
    const int S = (scount < (uint32_t)SORTN) ? (int)scount : SORTN;

    // ---- build 64-bit sort keys (pad with 0 => sorts last) ----
    for (int j = tid; j < SORTN; j += NTHREADS) {
        uint64_t key = 0ull;
        if (j < S) {
            uint32_t u  = candU[(size_t)b * SORTN + j];
            uint32_t ix = candIdx[(size_t)b * SORTN + j];
            key = ((uint64_t)u << 32) | (uint32_t)(~ix);  // desc score, asc idx
        }
        skey[j] = key;
    }
    __syncthreads();

    // ---- bitonic sort, descending ----
    for (int k = 2; k <= SORTN; k <<= 1) {
        for (int jj = k >> 1; jj > 0; jj >>= 1) {
            for (int i = tid; i < SORTN; i += NTHREADS) {
                int ixj = i ^ jj;
                if (ixj > i) {
                    uint64_t a = skey[i], c = skey[ixj];
                    bool up = ((i & k) == 0);
                    bool sw = up ? (a < c) : (a > c);   // descending overall
                    if (sw) { skey[i] = c; skey[ixj] = a; }
                }
            }
            __syncthreads();
        }
    }

    // ---- extract sorted original indices ----
    for (int j = tid; j < SORTN; j += NTHREADS)
        sidx[j] = (uint32_t)(~skey[j]);   // low 32 bits were ~idx
    __syncthreads();

    // ---- gather candidate boxes (AoS, 16B/slot) into LDS ----
    for (int j = tid; j < S; j += NTHREADS) {
        uint32_t oi = sidx[j];
        const float* gp = bbox + ((size_t)b * NMS_N + oi) * 4;
#if USE_ASYNC_GATHER && defined(__gfx1250__)
        uint32_t lds_off = (uint32_t)(uintptr_t)(&sbox[4 * j]);
        uint64_t gaddr   = (uint64_t)(uintptr_t)gp;
        asm volatile("global_load_async_to_lds_b128 %0, %1, off"
                     :: "v"(lds_off), "v"(gaddr) : "memory");
#else
        const float4 bb4 = *(const float4*)gp;
        sbox[4 * j + 0] = bb4.x; sbox[4 * j + 1] = bb4.y;
        sbox[4 * j + 2] = bb4.z; sbox[4 * j + 3] = bb4.w;
#endif
    }
#if USE_ASYNC_GATHER && defined(__gfx1250__)
    asm volatile("s_wait_asynccnt 0" ::: "memory");
#endif
    // ---- init suppression bitmask + control ----
    for (int w = tid; w < SORTN / 32; w += NTHREADS) ssup[w] = 0u;
    if (tid == 0) { sctl[0] = 0; sctl[1] = 0; sctl[2] = 0; }
    __syncthreads();

    // ---- greedy scan: thread 0 finds next survivor, block suppresses ----
    while (true) {
        if (tid == 0) {
            int p = sctl[1], kc = sctl[2], cur = -1;
            while (p < S && kc < NMS_K) {
                if (!((ssup[p >> 5] >> (p & 31)) & 1u)) {
                    float y1 = sbox[4*p], x1 = sbox[4*p+1];
                    float y2 = sbox[4*p+2], x2 = sbox[4*p+3];
                    if ((x2 - x1) * (y2 - y1) >= NMS_MINAREA) { cur = p; break; }
                }
                ++p;
            }
            if (cur >= 0) { skeep[kc] = cur; sctl[2] = kc + 1; sctl[1] = cur + 1; }
            else          { sctl[1] = p; }
            sctl[0] = cur;
        }
        __syncthreads();
        int cur = sctl[0];
        if (cur < 0) break;

        float iy1 = sbox[4*cur], ix1 = sbox[4*cur+1];
        float iy2 = sbox[4*cur+2], ix2 = sbox[4*cur+3];
        float iarea = (ix2 - ix1) * (iy2 - iy1);
        for (int j = cur + 1 + tid; j < S; j += NTHREADS) {
            if ((ssup[j >> 5] >> (j & 31)) & 1u) continue;
            float y1 = sbox[4*j], x1 = sbox[4*j+1];
            float y2 = sbox[4*j+2], x2 = sbox[4*j+3];
            float yy1 = fminf(fmaxf(y1, iy1), iy2);
            float xx1 = fminf(fmaxf(x1, ix1), ix2);
            float yy2 = fminf(fmaxf(y2, iy1), iy2);
            float xx2 = fminf(fmaxf(x2, ix1), ix2);
            float inter = (xx2 - xx1) * (yy2 - yy1);
            float areaj = (x2 - x1) * (y2 - y1);
            float iou = inter / (areaj + iarea - inter);
            if (iou > NMS_THRESH) atomicOr(&ssup[j >> 5], 1u << (j & 31));
        }
        __syncthreads();
    }
    __syncthreads();

    // ---- write outputs: boxes (B,K,4) | counts (B,1) | keeps (B,K) ----
    const int kc = sctl[2];
    const size_t OB = (size_t)NMS_B * NMS_K * 4;   // counts offset
    if (tid < NMS_K) {
        float b0 = 0.f, b1 = 0.f, b2 = 0.f, b3 = 0.f;
        int oi = 0;
        if (tid < kc) {
            int slot = skeep[tid];
            oi = (int)sidx[slot];
            b0 = sbox[4*slot]; b1 = sbox[4*slot+1];
            b2 = sbox[4*slot+2]; b3 = sbox[4*slot+3];
        }
        size_t base = ((size_t)b * NMS_K + tid) * 4;
        out[base + 0] = b0; out[base + 1] = b1;
        out[base + 2] = b2; out[base + 3] = b3;
        out[OB + NMS_B + (size_t)b * NMS_K + tid] = (float)oi;  // keeps as f32
    }
    if (tid == 0) out[OB + b] = (float)kc;                       // count as f32
}

// =====================================================================
// host launcher
// =====================================================================
extern "C" void kernel_launch(void* const* d_in, const int* in_sizes, int n_in,
                              void* d_out, int out_size, void* d_ws, size_t ws_size,
                              hipStream_t stream) {
    const float* scoress = (const float*)d_in[0];   // (B,N,2)
    const float* bboxess = (const float*)d_in[1];   // (B,N,4)
    float* out = (float*)d_out;                     // boxes | counts | keeps (flat f32)

    // workspace layout (~15 MB)
    uint32_t* uKeys    = (uint32_t*)d_ws;                              // B*N
    uint32_t* thr      = uKeys + (size_t)NMS_B * NMS_N;                // B
    uint32_t* selCount = thr + NMS_B;                                  // B
    uint32_t* candU    = selCount + NMS_B;                             // B*SORTN
    uint32_t* candIdx  = candU + (size_t)NMS_B * SORTN;                // B*SORTN

    const int total = NMS_B * NMS_N;
    score_key_kernel<<<(total + 255) / 256, 256, 0, stream>>>(
        (const float2*)scoress, uKeys, total);
    radix_select_kernel<<<NMS_B, NTHREADS, 0, stream>>>(uKeys, thr, selCount);
    compact_kernel<<<(total + 255) / 256, 256, 0, stream>>>(
        uKeys, thr, selCount, candU, candIdx);
    nms_kernel<<<NMS_B, NTHREADS, SM_TOTAL, stream>>>(
        bboxess, candU, candIdx, selCount, out);
}